// HybridMemory_85023172592064
// MI455X (gfx1250) — compile-verified
//
#include <hip/hip_runtime.h>
#include <hip/hip_bf16.h>
#include <math.h>

// Problem constants (match reference)
#define BB 1024      // batch
#define FF 256       // feature dim
#define NN 65536     // bank rows
#define CC 1000      // classes
#define CPAD 1008    // classes padded to 16
#define TEMP 0.05f
#define MOMENTUM 0.2f
#define IGNORE_IDX 750  // SOURCE_CLASSES - 1

typedef __attribute__((ext_vector_type(2))) float v2f;
typedef __attribute__((ext_vector_type(8))) float v8f;

// ---- workspace layout (float indices) ----
// [0, 258048)            classfeat  [CPAD x FF]   (zeroed)
// [258048, 259056)       numsf      [CPAD]        (zeroed)
// [259056, 259058)       accum      [2]           (zeroed)
// [259072, 521216)       x          [BB x FF]
// [521216, 1553408)      E (exps)   [CPAD x BB]
// [1553408, 1554432)     idx (int)  [BB]
#define WS_CF    0
#define WS_NUMS  258048
#define WS_ACC   259056
#define WS_ZEND  259072
#define WS_X     259072
#define WS_E     521216
#define WS_IDX   1553408

// -------------------- zero init --------------------
__global__ void k_zero(float* ws) {
    int i = blockIdx.x * blockDim.x + threadIdx.x;
    if (i < WS_ZEND) ws[i] = 0.0f;
}

// -------------------- normalize inputs, remap idx --------------------
__global__ void k_normalize(const float* __restrict__ inputs,
                            const int* __restrict__ indexes,
                            float* __restrict__ x, int* __restrict__ idx) {
    __shared__ float sred[FF];
    int b = blockIdx.x, t = threadIdx.x;
    float v = inputs[b * FF + t];
    sred[t] = v * v;
    __syncthreads();
    for (int s = FF / 2; s > 0; s >>= 1) {
        if (t < s) sred[t] += sred[t + s];
        __syncthreads();
    }
    float nrm = sqrtf(sred[0]);
    x[b * FF + t] = v / nrm;
    if (t == 0) {
        int i = indexes[b] - 1;
        if (i == 5554) i = IGNORE_IDX;
        idx[b] = i;
    }
}

// -------------------- per-class feature sums + counts --------------------
__global__ void k_classsum(const float* __restrict__ features,
                           const int* __restrict__ labels,
                           float* __restrict__ classfeat,
                           float* __restrict__ numsf) {
    int n = blockIdx.x, t = threadIdx.x;
    int c = labels[n];
    atomicAdd(&classfeat[c * FF + t], features[n * FF + t]);
    if (t == 0) atomicAdd(&numsf[c], 1.0f);
}

// -------------------- WMMA f32 GEMM: E = exp(classfeat @ x^T /(T*nums)) * mask
// A = classfeat [CPAD x FF] (M=class, K=feature)
// B = x         [BB   x FF] (N=batch, K=feature)  -> B matrix is K x N
// One wave per 16x16 tile; K stepped by 4 via V_WMMA_F32_16X16X4_F32.
__global__ void k_wmma_exps(const float* __restrict__ classfeat,
                            const float* __restrict__ numsf,
                            const float* __restrict__ x,
                            float* __restrict__ E) {
    const int cbase = blockIdx.x * 16;
    const int bbase = blockIdx.y * 16;
    const int lane  = threadIdx.x;      // 0..31 (wave32)
    const int m     = lane & 15;        // M for A / N for B / N col for D
    const int g     = lane >> 4;        // lane-group

    const float* arow = classfeat + (cbase + m) * FF + 2 * g;  // K = 2g + {0,1}
    const float* brow = x + (bbase + m) * FF + 2 * g;

    v8f acc = {};
#pragma unroll
    for (int k0 = 0; k0 < FF; k0 += 4) {
        v2f a = *(const v2f*)(arow + k0);
        v2f b = *(const v2f*)(brow + k0);
        acc = __builtin_amdgcn_wmma_f32_16x16x4_f32(
            /*neg_a=*/false, a, /*neg_b=*/false, b,
            /*c_mod=*/(short)0, acc, /*reuse_a=*/false, /*reuse_b=*/false);
    }

    // D layout: VGPR r -> class row cbase + r + 8*g, batch col bbase + m
    const int bcol = bbase + m;
#pragma unroll
    for (int r = 0; r < 8; ++r) {
        int c = cbase + r + 8 * g;
        if (c < CC) {
            float n = numsf[c];
            float val = (n > 0.0f) ? __expf(acc[r] / (TEMP * n)) : 0.0f;
            E[c * BB + bcol] = val;
        }
    }
}

// -------------------- per-batch softmax denom + NLL --------------------
__global__ void k_loss_reduce(const float* __restrict__ E,
                              const int* __restrict__ idx,
                              const int* __restrict__ labels,
                              float* __restrict__ accum) {
    int b = blockIdx.x * blockDim.x + threadIdx.x;  // 0..1023
    if (b >= BB) return;
    float S = 0.0f;
    for (int c = 0; c < CC; ++c) S += E[c * BB + b];   // coalesced across threads
    int t = labels[idx[b]];
    float p = E[t * BB + b] / (S + 1e-6f);
    float nll = -logf(p + 1e-6f);
    float valid = (t != IGNORE_IDX) ? 1.0f : 0.0f;
    atomicAdd(&accum[0], nll * valid);
    atomicAdd(&accum[1], valid);
}

__global__ void k_finalize(const float* __restrict__ accum, float* __restrict__ out) {
    out[0] = accum[0] / fmaxf(accum[1], 1.0f);
}

// -------------------- sequential momentum scan (order-dependent) --------------------
__global__ void k_scan(const float* __restrict__ x,
                       const int* __restrict__ idx,
                       float* __restrict__ nf) {
    __shared__ float sred[FF];
    int t = threadIdx.x;
    for (int i = 0; i < BB; ++i) {
        int yi = idx[i];
        float v = MOMENTUM * nf[yi * FF + t] + (1.0f - MOMENTUM) * x[i * FF + t];
        sred[t] = v * v;
        __syncthreads();
        for (int s = FF / 2; s > 0; s >>= 1) {
            if (t < s) sred[t] += sred[t + s];
            __syncthreads();
        }
        v /= sqrtf(sred[0]);
        nf[yi * FF + t] = v;
        __syncthreads();
    }
}

extern "C" void kernel_launch(void* const* d_in, const int* in_sizes, int n_in,
                              void* d_out, int out_size, void* d_ws, size_t ws_size,
                              hipStream_t stream) {
    const float* inputs   = (const float*)d_in[0];   // [B,F]
    const int*   indexes  = (const int*)d_in[1];     // [B]
    const float* features = (const float*)d_in[2];   // [N,F]
    const int*   labels   = (const int*)d_in[3];     // [N]

    float* ws = (float*)d_ws;
    float* classfeat = ws + WS_CF;
    float* numsf     = ws + WS_NUMS;
    float* accum     = ws + WS_ACC;
    float* x         = ws + WS_X;
    float* E         = ws + WS_E;
    int*   idx       = (int*)(ws + WS_IDX);

    float* out_loss = (float*)d_out;          // d_out[0]
    float* new_feat = (float*)d_out + 1;      // d_out[1 .. N*F]

    // 1) zero classfeat / nums / accumulators
    k_zero<<<(WS_ZEND + 255) / 256, 256, 0, stream>>>(ws);

    // 2) x = normalize(inputs); idx remap
    k_normalize<<<BB, FF, 0, stream>>>(inputs, indexes, x, idx);

    // 3) per-class feature sums + counts (streams the 64MB bank once)
    k_classsum<<<NN, FF, 0, stream>>>(features, labels, classfeat, numsf);

    // 4) fused WMMA GEMM + mask/mean/exp epilogue
    dim3 gemm_grid(CPAD / 16, BB / 16);
    k_wmma_exps<<<gemm_grid, 32, 0, stream>>>(classfeat, numsf, x, E);

    // 5) softmax denominator + NLL, accumulate loss
    k_loss_reduce<<<BB / 256, 256, 0, stream>>>(E, idx, labels, accum);
    k_finalize<<<1, 1, 0, stream>>>(accum, out_loss);

    // 6) new_features = features, then order-preserving momentum scan
    hipMemcpyAsync(new_feat, features, (size_t)NN * FF * sizeof(float),
                   hipMemcpyDeviceToDevice, stream);
    k_scan<<<1, FF, 0, stream>>>(x, idx, new_feat);
}